// PatchAttacker_14525579395105
// MI455X (gfx1250) — compile-verified
//
#include <hip/hip_runtime.h>
#include <stdint.h>

#define BB 16
#define NN 16
#define HH 512
#define WW 512
#define PP 512
#define EPSF 1e-6f

// ---------------------------------------------------------------------------
// Kernel 0: zero the 6-float stats accumulator in workspace
// ---------------------------------------------------------------------------
__global__ void zero_ws_kernel(float* ws) {
    if (threadIdx.x < 8) ws[threadIdx.x] = 0.0f;
}

// ---------------------------------------------------------------------------
// Kernel 1: per-channel sum and sum-of-squares of the 512x512x3 patch.
// Block-level LDS tree reduce, then 6 float atomics per block into ws[0..5].
// ---------------------------------------------------------------------------
__global__ void patch_stats_kernel(const float* __restrict__ patch,
                                   float* __restrict__ ws) {
    float s[3] = {0.f, 0.f, 0.f};
    float q[3] = {0.f, 0.f, 0.f};
    const int stride = gridDim.x * blockDim.x;
    for (int i = blockIdx.x * blockDim.x + threadIdx.x; i < PP * PP; i += stride) {
        const float* p = patch + (size_t)i * 3;
#pragma unroll
        for (int c = 0; c < 3; ++c) {
            float v = p[c];
            s[c] += v;
            q[c] += v * v;
        }
    }
    __shared__ float red[6 * 256];
    const int tid = threadIdx.x;
#pragma unroll
    for (int c = 0; c < 3; ++c) {
        red[c * 256 + tid]       = s[c];
        red[(3 + c) * 256 + tid] = q[c];
    }
    __syncthreads();
    for (int off = 128; off > 0; off >>= 1) {
        if (tid < off) {
#pragma unroll
            for (int k = 0; k < 6; ++k)
                red[k * 256 + tid] += red[k * 256 + tid + off];
        }
        __syncthreads();
    }
    if (tid == 0) {
#pragma unroll
        for (int k = 0; k < 6; ++k) atomicAdd(&ws[k], red[k * 256]);
    }
}

// ---------------------------------------------------------------------------
// Kernel 2: bulk copy images -> out via CDNA5 async global<->LDS DMA.
// Each block moves 4 tiles of 256 float4 (16 KB). One asm block issues all
// four async loads (in-order on ASYNCcnt), drains with a single
// s_wait_asynccnt 0, then issues the four async stores and drains once more.
// The async ops add INST_OFFSET to BOTH the LDS and global addresses, so a
// single LDS base + offset:4096*k addresses the matching stage slot.
// Data never passes through VGPRs.
// ---------------------------------------------------------------------------
__global__ void async_copy_kernel(const float4* __restrict__ src,
                                  float4* __restrict__ dst, int n) {
    __shared__ __align__(16) float4 stage[4 * 256];  // 16 KB, 4 tile slots
    const int tid = threadIdx.x;
    const long base = (long)blockIdx.x * (4 * 256) + tid;
    if (base >= n) return;
    // low 32 bits of a generic pointer into LDS == LDS byte address
    uint32_t lds_addr = (uint32_t)(uintptr_t)&stage[tid];
    const float4* sp = src + base;  // tile 0; tiles k at +4096*k bytes
    float4* dp = dst + base;
    asm volatile(
        "global_load_async_to_lds_b128 %0, %1, off\n\t"
        "global_load_async_to_lds_b128 %0, %1, off offset:4096\n\t"
        "global_load_async_to_lds_b128 %0, %1, off offset:8192\n\t"
        "global_load_async_to_lds_b128 %0, %1, off offset:12288\n\t"
        "s_wait_asynccnt 0x0\n\t"
        "global_store_async_from_lds_b128 %2, %0, off\n\t"
        "global_store_async_from_lds_b128 %2, %0, off offset:4096\n\t"
        "global_store_async_from_lds_b128 %2, %0, off offset:8192\n\t"
        "global_store_async_from_lds_b128 %2, %0, off offset:12288\n\t"
        "s_wait_asynccnt 0x0"
        :
        : "v"(lds_addr), "v"(sp), "v"(dp)
        : "memory");
}

// ---------------------------------------------------------------------------
// Kernel 3: one workgroup per image; serial chain of 16 patch applications.
// Wave-per-row / lane-per-column mapping: no integer division, and all
// y-direction bilinear coefficients are wave-uniform (computed once per row).
// ---------------------------------------------------------------------------
__global__ void apply_patches_kernel(float* __restrict__ out,
                                     const float* __restrict__ boxes,
                                     const float* __restrict__ patch,
                                     const float* __restrict__ ws) {
    const int b    = blockIdx.x;
    const int tid  = threadIdx.x;
    const int lane = tid & 31;
    const int wave = tid >> 5;  // 0..7

    // patch channel stats (uniform across block)
    const float invN = 1.0f / (float)(PP * PP);
    float pm[3], psd[3];
#pragma unroll
    for (int c = 0; c < 3; ++c) {
        float m   = ws[c] * invN;
        float var = fmaxf(ws[3 + c] * invN - m * m, 0.0f);
        pm[c]  = m;
        psd[c] = sqrtf(var) + EPSF;
    }

    __shared__ float red[6 * 256];
    __shared__ float bg[6];
    float* img = out + (size_t)b * HH * WW * 3;

    for (int n = 0; n < NN; ++n) {
        const float* bx = boxes + ((size_t)b * NN + n) * 4;
        float ymin = bx[0], xmin = bx[1], ymax = bx[2], xmax = bx[3];
        float h  = ymax - ymin, w = xmax - xmin;
        float pw = h * 0.3f;        // SCALE
        float ph = 1.0f * pw;       // ASPECT * pw
        float oy = ymin + h * 0.5f, ox = xmin + w * 0.5f;
        float yp = fmaxf(oy - ph * 0.5f, 0.0f);
        float xp = fmaxf(ox - pw * 0.5f, 0.0f);
        if (yp + ph > (float)HH) yp = (float)HH - ph;
        if (xp + pw > (float)WW) xp = (float)WW - pw;
        if (!(ph > 60.0f)) continue;  // MIN_PATCH_H filter: masked write -> no-op

        const int y0  = (int)yp;      // tf.cast truncation
        const int x0  = (int)xp;
        const int phi = (int)ph;
        const int pwi = (int)pw;

        // ---- pass 1: background mean/std over current rectangle ----
        float s0 = 0.f, s1 = 0.f, s2 = 0.f, q0 = 0.f, q1 = 0.f, q2 = 0.f;
        for (int ry = wave; ry < phi; ry += 8) {
            const float* row = img + ((size_t)(y0 + ry) * WW + x0) * 3;
            for (int rx = lane; rx < pwi; rx += 32) {
                const float* p = row + rx * 3;
                float r = p[0], g = p[1], bl = p[2];
                s0 += r;     s1 += g;     s2 += bl;
                q0 += r * r; q1 += g * g; q2 += bl * bl;
            }
        }
        red[0 * 256 + tid] = s0; red[1 * 256 + tid] = s1; red[2 * 256 + tid] = s2;
        red[3 * 256 + tid] = q0; red[4 * 256 + tid] = q1; red[5 * 256 + tid] = q2;
        __syncthreads();
        for (int off = 128; off > 0; off >>= 1) {
            if (tid < off) {
#pragma unroll
                for (int k = 0; k < 6; ++k)
                    red[k * 256 + tid] += red[k * 256 + tid + off];
            }
            __syncthreads();
        }
        if (tid == 0) {
            float cnt = fmaxf((float)(phi * pwi), 1.0f);
#pragma unroll
            for (int c = 0; c < 3; ++c) {
                float m   = red[c * 256] / cnt;
                float var = fmaxf(red[(3 + c) * 256] / cnt - m * m, 0.0f);
                bg[c]     = m;
                bg[3 + c] = sqrtf(var) + EPSF;
            }
        }
        __syncthreads();

        // fused moment match (bilinear is linear, so sample-then-affine
        // equals the reference's affine-then-sample): val = a[c]*s + c[c]
        const float a0 = bg[3] / psd[0], a1 = bg[4] / psd[1], a2 = bg[5] / psd[2];
        const float c0 = bg[0] - pm[0] * a0;
        const float c1 = bg[1] - pm[1] * a1;
        const float c2 = bg[2] - pm[2] * a2;

        const float phf = fmaxf((float)phi, 1.0f);
        const float pwf = fmaxf((float)pwi, 1.0f);
        const float scy = (float)PP / phf;
        const float scx = (float)PP / pwf;

        // ---- pass 2: bilinear resample patch into rectangle ----
        for (int ry = wave; ry < phi; ry += 8) {
            // wave-uniform y coefficients
            float sy = fminf(fmaxf(((float)ry + 0.5f) * scy - 0.5f, 0.0f),
                             (float)(PP - 1));
            int   iy0 = (int)sy;
            int   iy1 = min(iy0 + 1, PP - 1);
            float wy  = sy - (float)iy0;
            const float* prow0 = patch + (size_t)iy0 * PP * 3;
            const float* prow1 = patch + (size_t)iy1 * PP * 3;
            float* orow = img + ((size_t)(y0 + ry) * WW + x0) * 3;
            for (int rx = lane; rx < pwi; rx += 32) {
                float sx = fminf(fmaxf(((float)rx + 0.5f) * scx - 0.5f, 0.0f),
                                 (float)(PP - 1));
                int   ix0 = (int)sx;
                int   ix1 = min(ix0 + 1, PP - 1);
                float wx  = sx - (float)ix0;
                const float* p00 = prow0 + ix0 * 3;
                const float* p01 = prow0 + ix1 * 3;
                const float* p10 = prow1 + ix0 * 3;
                const float* p11 = prow1 + ix1 * 3;
                float* q = orow + rx * 3;
                const float av[3] = {a0, a1, a2};
                const float cv[3] = {c0, c1, c2};
#pragma unroll
                for (int c = 0; c < 3; ++c) {
                    float top  = p00[c] + wx * (p01[c] - p00[c]);
                    float bot  = p10[c] + wx * (p11[c] - p10[c]);
                    float samp = top + wy * (bot - top);
                    q[c] = av[c] * samp + cv[c];
                }
            }
        }
        // next iteration's stats (same workgroup) must see these writes
        __threadfence_block();
        __syncthreads();
    }
}

// ---------------------------------------------------------------------------
extern "C" void kernel_launch(void* const* d_in, const int* in_sizes, int n_in,
                              void* d_out, int out_size, void* d_ws, size_t ws_size,
                              hipStream_t stream) {
    const float* images = (const float*)d_in[0];  // [16,512,512,3]
    const float* boxes  = (const float*)d_in[1];  // [16,16,4]
    const float* patch  = (const float*)d_in[2];  // [512,512,3]
    float* out = (float*)d_out;                   // [16,512,512,3]
    float* ws  = (float*)d_ws;                    // stats: sum[3], sumsq[3]

    zero_ws_kernel<<<1, 32, 0, stream>>>(ws);
    patch_stats_kernel<<<384, 256, 0, stream>>>(patch, ws);

    const int n4 = BB * HH * WW * 3 / 4;  // 3,145,728 float4, 1024 per block
    async_copy_kernel<<<n4 / (4 * 256), 256, 0, stream>>>(
        (const float4*)images, (float4*)out, n4);

    apply_patches_kernel<<<BB, 256, 0, stream>>>(out, boxes, patch, ws);
}